// SigmoidLoss_34230889349773
// MI455X (gfx1250) — compile-verified
//
#include <hip/hip_runtime.h>
#include <hip/hip_bf16.h>

// ---------------------------------------------------------------------------
// SigmoidLoss: per-row masked max of log(clip(sigmoid(x))), weighted mean.
// Memory-bound streaming reduction: 655 MB @ 23.3 TB/s => ~28 us floor.
// Kernel 1: direct 128-bit NT global loads (no reuse -> no LDS/TDM staging),
//           wave32 shuffle + LDS reduction, one transcendental per row.
// Kernel 2: deterministic final sum of 16384 row losses via
//           V_WMMA_F32_16X16X4_F32 (A = ones => D accumulates chunk sums).
// ---------------------------------------------------------------------------

typedef __attribute__((ext_vector_type(2))) float v2f;
typedef __attribute__((ext_vector_type(4))) float v4f;
typedef __attribute__((ext_vector_type(8))) float v8f;
typedef __attribute__((ext_vector_type(4))) int   v4i;

#define BLK 256
#define SENT (-3.0e38f)
#define EPSV (1e-6f)

__global__ __launch_bounds__(BLK)
void SigmoidLoss_rowmax_kernel(const float* __restrict__ inp,
                               const int*   __restrict__ tgt,
                               const float* __restrict__ w,
                               float*       __restrict__ row_loss,
                               int C) {
    const int row   = blockIdx.x;
    const size_t base = (size_t)row * (size_t)C;
    const int C4 = C >> 2;
    const v4f* in4 = (const v4f*)(inp + base);
    const v4i* tg4 = (const v4i*)(tgt + base);

    // Per-lane masked max over the row (streaming, non-temporal).
    float m = SENT;
    for (int j = threadIdx.x; j < C4; j += BLK) {
        v4f x = __builtin_nontemporal_load(in4 + j);
        v4i t = __builtin_nontemporal_load(tg4 + j);
        m = fmaxf(m, t.x > 0 ? x.x : SENT);
        m = fmaxf(m, t.y > 0 ? x.y : SENT);
        m = fmaxf(m, t.z > 0 ? x.z : SENT);
        m = fmaxf(m, t.w > 0 ? x.w : SENT);
    }
    for (int j = (C4 << 2) + threadIdx.x; j < C; j += BLK) {
        float x = inp[base + j];
        int   t = tgt[base + j];
        m = fmaxf(m, t > 0 ? x : SENT);
    }

    // wave32 xor-shuffle max reduction
    #pragma unroll
    for (int off = 16; off >= 1; off >>= 1)
        m = fmaxf(m, __shfl_xor(m, off, 32));

    __shared__ float smax[BLK / 32];
    const int lane = threadIdx.x & 31;
    const int wid  = threadIdx.x >> 5;
    if (lane == 0) smax[wid] = m;
    __syncthreads();

    if (threadIdx.x == 0) {
        float mm = smax[0];
        #pragma unroll
        for (int i = 1; i < BLK / 32; ++i) mm = fmaxf(mm, smax[i]);
        float loss = 0.0f;
        if (mm > -1.0e38f) {   // row had at least one positive label
            // log(clip(sigmoid(m), eps, 1-eps)); monotone => max commutes
            float s = 1.0f / (1.0f + expf(-mm));
            s = fminf(fmaxf(s, EPSV), 1.0f - EPSV);
            loss = -logf(s);
        }
        row_loss[row] = loss * w[row];
    }
}

// Final reduction with WMMA: D = ones(16x4) * B(4x16) + C accumulates column
// sums of each 64-float chunk; every element lands in exactly one column, so
// sum of D row 0 (lanes 0..15 of c[0]) == sum of the chunk, independent of
// the exact B lane layout. Single wave32, EXEC all ones, fully deterministic.
__global__ __launch_bounds__(32)
void SigmoidLoss_reduce_kernel(const float* __restrict__ v,
                               float*       __restrict__ out,
                               int n, float inv_n) {
    const int lane = threadIdx.x;
    v8f c = {};
    v2f ones; ones.x = 1.0f; ones.y = 1.0f;

    const int chunks = n >> 6;                 // 64 floats per WMMA
    const v2f* v2 = (const v2f*)v;
    for (int ch = 0; ch < chunks; ++ch) {
        v2f b = v2[(ch << 5) + lane];          // 32 lanes * 2 floats = 64
        c = __builtin_amdgcn_wmma_f32_16x16x4_f32(
                /*neg_a=*/false, ones, /*neg_b=*/false, b,
                /*c_mod=*/(short)0, c, /*reuse_a=*/false, /*reuse_b=*/false);
    }

    // Scalar tail (n not multiple of 64): reduce across all 32 lanes.
    float t = 0.0f;
    for (int i = (chunks << 6) + lane; i < n; i += 32) t += v[i];
    #pragma unroll
    for (int off = 16; off >= 1; off >>= 1) t += __shfl_xor(t, off, 32);

    // Sum D row M=0: lanes 0..15 of c[0] hold N=0..15 column sums.
    float s = c[0];
    s += __shfl_xor(s, 1, 32);
    s += __shfl_xor(s, 2, 32);
    s += __shfl_xor(s, 4, 32);
    s += __shfl_xor(s, 8, 32);

    if (lane == 0) out[0] = (s + t) * inv_n;
}

extern "C" void kernel_launch(void* const* d_in, const int* in_sizes, int n_in,
                              void* d_out, int out_size, void* d_ws, size_t ws_size,
                              hipStream_t stream) {
    const float* inp = (const float*)d_in[0];
    const int*   tgt = (const int*)d_in[1];
    const float* w   = (const float*)d_in[2];
    float*       out = (float*)d_out;

    const int B = in_sizes[2];                       // 16384
    const int C = (int)((long long)in_sizes[0] / B); // 5000
    float* row_loss = (float*)d_ws;                  // B floats of scratch

    SigmoidLoss_rowmax_kernel<<<B, BLK, 0, stream>>>(inp, tgt, w, row_loss, C);
    SigmoidLoss_reduce_kernel<<<1, 32, 0, stream>>>(row_loss, out, B,
                                                    1.0f / (float)B);
}